// Embedding_73572789780642
// MI455X (gfx1250) — compile-verified
//
#include <hip/hip_runtime.h>
#include <hip/hip_bf16.h>
#include <stddef.h>

// Embedding gather: out[t, 0:1024] = weight[ids[t], 0:1024], fp32.
// Pure bandwidth problem: ~98MB weight reads (fits L2, keep RT) +
// 128MB streamed output (write-once, use non-temporal stores).
// One 4KB row == 256 lanes x float4 -> one token per block-iteration.

typedef float v4f __attribute__((ext_vector_type(4)));

#define D_MODEL 1024
#define BLOCK   256   // 8 wave32s; 256 * 4 floats = 1024 = one full row

__global__ __launch_bounds__(BLOCK) void embed_gather_kernel(
    const int* __restrict__ ids,      // [n_tokens]
    const float* __restrict__ weight, // [VOCAB, D_MODEL]
    float* __restrict__ out,          // [n_tokens, D_MODEL]
    int n_tokens)
{
    const int col = threadIdx.x * 4;          // this lane's float4 column
    const int stride = gridDim.x;

    for (int t = blockIdx.x; t < n_tokens; t += stride) {
        // Block-uniform row index -> scalar load path.
        const int row = ids[t];

        // Prefetch the NEXT token's row into cache while we stream this one.
        // Lowers to global_prefetch_b8 (gfx1250); each lane covers its 16B
        // slice, so the whole 4KB row (32 cachelines) gets touched.
        const int tn = t + stride;
        if (tn < n_tokens) {
            const int rown = ids[tn];
            __builtin_prefetch(weight + (size_t)rown * D_MODEL + col, /*rw=*/0, /*loc=*/3);
        }

        // Coalesced 16B load of this lane's slice of the row (regular-temporal:
        // repeated tokens re-hit these lines in the 192MB L2).
        const v4f* src = reinterpret_cast<const v4f*>(weight + (size_t)row * D_MODEL + col);
        v4f v = *src;

        // Output is written exactly once and never re-read by this kernel:
        // non-temporal store (TH=NT) keeps the 128MB write stream from
        // evicting the hot weight rows out of L2.
        v4f* dst = reinterpret_cast<v4f*>(out + (size_t)t * D_MODEL + col);
        __builtin_nontemporal_store(v, dst);
    }
}

extern "C" void kernel_launch(void* const* d_in, const int* in_sizes, int n_in,
                              void* d_out, int out_size, void* d_ws, size_t ws_size,
                              hipStream_t stream)
{
    const int*   ids    = (const int*)d_in[0];     // token_ids: 8*4096 int32
    const float* weight = (const float*)d_in[1];   // [50257, 1024] fp32
    float*       out    = (float*)d_out;           // [8, 4096, 1024] fp32

    const int n_tokens = in_sizes[0];              // 32768

    // Grid-stride so each block has a "next token" to prefetch; 4096 blocks
    // is plenty to saturate every WGP while keeping ~8 tokens per block.
    int grid = 4096;
    if (grid > n_tokens) grid = n_tokens;
    if (grid < 1) grid = 1;

    embed_gather_kernel<<<grid, BLOCK, 0, stream>>>(ids, weight, out, n_tokens);
}